// BatchBlur_SV3_4105988735745
// MI455X (gfx1250) — compile-verified
//
#include <hip/hip_runtime.h>

// BatchBlur (per-pixel 19x19 kernel, reflect pad, /361) for MI455X gfx1250.
// HBM floor: 189 MB single-use weight stream @ 23.3 TB/s => ~8.1 us.
// Round-4: the input gathers (36/pixel, ~4 cachelines each) were the binding
// VMEM-pipe cost. Each block's 32 consecutive pixels share one reflected
// 3x19x50 patch (11.4 KB) -> stage it in LDS once (coalesced, reflection
// resolved at stage time), inner loop reads LDS (bank-conflict-free).
// VMEM txns drop ~8x; no boundary path remains. Weight stream: 12 NT b32
// coalesced loads/pixel. Reduction: v_wmma_f32_16x16x4_f32 per channel.

typedef float v2f __attribute__((ext_vector_type(2)));
typedef float v8f __attribute__((ext_vector_type(8)));

#define HH    256
#define WW    256
#define CC    3
#define LL    19
#define K2    361            // 19*19
#define PADR  9              // L/2
#define PLANE (HH * WW)      // 65536
#define NPIX  (2 * PLANE)    // B*H*W = 131072
#define NP    4              // pixels per wave
#define BPX   32             // pixels per block (8 waves * NP, divides 256)

#define PROWS 19
#define PCOLS 50             // 32 pixels + 18 halo
#define PPLANE (PROWS * PCOLS)        // 950 floats per channel
#define PATCH_ELEMS (CC * PPLANE)     // 2850 floats = 11.4 KB

// reflect (mirror, no edge repeat): q in [-9, 266] -> [0,255]
__device__ __forceinline__ int reflect_idx(int q) {
    int a = (q < 0) ? -q : q;
    int m = 2 * (HH - 1) - a;
    return (a < m) ? a : m;
}

// Sum acc over all 32 lanes with one f32 WMMA.
// A (16x4): a.x = acc -> lanes 0..15 = A[m,0], lanes 16..31 = A[m,2]; a.y = 0.
// B = ones (layout-independent) => D[m,n] = acc[m] + acc[m+16] = partial[m].
// D VGPR v: lanes 0..15 = partial[v], lanes 16..31 = partial[v+8];
// sum of 8 D VGPRs gives half-sums, xor-16 shuffle finishes.
__device__ __forceinline__ float wave_reduce_wmma(float acc) {
    v2f a; a.x = acc;  a.y = 0.0f;
    v2f b; b.x = 1.0f; b.y = 1.0f;
    v8f c = {};
    v8f d = __builtin_amdgcn_wmma_f32_16x16x4_f32(
        false, a, false, b, (short)0, c, false, false);
    float s = d[0] + d[1] + d[2] + d[3] + d[4] + d[5] + d[6] + d[7];
    s += __shfl_xor(s, 16, 32);
    return s;
}

__global__ __launch_bounds__(256) void batchblur_sv3_kernel(
        const float* __restrict__ inp,    // (2, 3, 256, 256) f32
        const float* __restrict__ kern,   // (2, 65536, 19, 19) f32
        float* __restrict__ out) {        // (2, 3, 256, 256) f32
    __shared__ float patch[PATCH_ELEMS];

    const int tid  = threadIdx.x;
    const int lane = tid & 31;
    const int wid  = __builtin_amdgcn_readfirstlane(tid >> 5);  // scalar wave id

    const int pidblk = (int)blockIdx.x * BPX;   // first pixel of the block
    const int b    = pidblk >> 16;
    const int p0   = pidblk & 0xFFFF;
    const int h    = p0 >> 8;                   // whole block in one row
    const int wblk = p0 & 0xFF;                 // multiple of 32

    // ---- stage reflected 3x19x50 patch into LDS (coalesced rows) ----
    const float* bplane = inp + (size_t)b * (CC * PLANE);
#pragma unroll
    for (int base = 0; base < PATCH_ELEMS; base += 256) {
        int idx = base + tid;
        if (idx < PATCH_ELEMS) {
            int c   = idx / PPLANE;
            int rem = idx - c * PPLANE;
            int r   = rem / PCOLS;
            int col = rem - r * PCOLS;
            int gy  = reflect_idx(h - PADR + r);
            int gx  = reflect_idx(wblk - PADR + col);
            patch[idx] = bplane[(c << 16) + (gy << 8) + gx];
        }
    }
    __syncthreads();

    // ---- per-lane LDS tap offsets: lane-pure, shared by all pixels ----
    int loffb[12];
    {
        int ki = (lane >= LL) ? 1 : 0;          // k = lane initially
        int kj = lane - LL * ki;
#pragma unroll
        for (int it = 0; it < 12; ++it) {
            int off = ki * PCOLS + kj;
            if (it == 11)                        // clamp phantom taps (k>360)
                off = (lane >= 9) ? (18 * PCOLS + 18) : off;
            loffb[it] = off << 2;
            ki += 1; kj += 13;                   // k += 32 = 1*19 + 13
            if (kj >= LL) { kj -= LL; ki += 1; }
        }
    }

    const int pid0 = pidblk + wid * NP;          // this wave's first pixel
    const float* kp0 = kern + (size_t)pid0 * K2; // SGPR base
    float* op0 = out + (size_t)b * (CC * PLANE) + (p0 + wid * NP);
    const char* pbase = (const char*)patch + ((wid * NP) << 2);

    // ---- pixel loop (rolled: one pixel's weights live at a time) ----
#pragma unroll 1
    for (int q = 0; q < NP; ++q) {
        // stream this pixel's 361 weights (coalesced 128B/wave, NT)
        const float* kp = kp0 + (size_t)q * K2;
        float wreg[12];
#pragma unroll
        for (int it = 0; it < 12; ++it) {
            int k  = it * 32 + lane;
            int kk = (k < K2) ? k : (K2 - 1);
            float v = __builtin_nontemporal_load(kp + kk);
            wreg[it] = (k < K2) ? v : 0.0f;      // zero phantom taps
        }

        // inner loop: 3 ds_load + 3 FMA per iteration (imm channel offsets)
        const char* pb = pbase + (q << 2);
        float acc0 = 0.f, acc1 = 0.f, acc2 = 0.f;
#pragma unroll
        for (int it = 0; it < 12; ++it) {
            const float* px = (const float*)(pb + loffb[it]);
            const float wv = wreg[it];
            acc0 += wv * px[0];                  // c=0
            acc1 += wv * px[PPLANE];             // c=1 (imm 3800B)
            acc2 += wv * px[2 * PPLANE];         // c=2 (imm 7600B)
        }

        // 32-lane reductions via WMMA (EXEC all ones here)
        const float t0 = wave_reduce_wmma(acc0);
        const float t1 = wave_reduce_wmma(acc1);
        const float t2 = wave_reduce_wmma(acc2);

        if (lane == 0) {
            const float s = 1.0f / (float)K2;
            op0[q]             = t0 * s;
            op0[q + PLANE]     = t1 * s;
            op0[q + 2 * PLANE] = t2 * s;
        }
    }
}

extern "C" void kernel_launch(void* const* d_in, const int* in_sizes, int n_in,
                              void* d_out, int out_size, void* d_ws, size_t ws_size,
                              hipStream_t stream) {
    const float* inp  = (const float*)d_in[0];   // (2,3,256,256)
    const float* kern = (const float*)d_in[1];   // (2,65536,19,19)
    float* out = (float*)d_out;                  // (2,3,256,256)

    const int blocks = NPIX / BPX;               // 4096 blocks of 8 waves
    batchblur_sv3_kernel<<<dim3(blocks), dim3(256), 0, stream>>>(inp, kern, out);
}